// GruDirection3d_37512244363989
// MI455X (gfx1250) — compile-verified
//
#include <hip/hip_runtime.h>
#include <cstdint>

// (B,C,D,H,W) = (1,32,64,128,128) fp32 backward scan along D:
//   out[d] = z[d]*_h[d] + (1-z[d])*out[d+1],  out[D] := h0
// Memory-bound: ~386 MB -> ~16.6 us floor at 23.3 TB/s.

#define KD   64           // depth
#define KHW4 4096         // float4 elements per (c,d) slice (128*128/4)
#define KC   32           // B*C channels
#define KCOLS (KC * KHW4) // 131072 float4 columns

typedef __attribute__((ext_vector_type(4))) float v4f;

// ---- CDNA5 async-copy (global -> LDS, ASYNCcnt-tracked) ----------------
#if defined(__HIP_DEVICE_COMPILE__) && \
    __has_builtin(__builtin_amdgcn_global_load_async_to_lds_b128)
#define USE_ASYNC 1
#else
#define USE_ASYNC 0
#endif

#if USE_ASYNC
// Builtin expects pointers to 128-bit int vectors (gcc vector_size style),
// global side in AS1, LDS side in AS3, both non-const (per clang diagnostic).
typedef int v4i_gcc __attribute__((vector_size(16)));
typedef __attribute__((address_space(1))) v4i_gcc as1_v4i;
typedef __attribute__((address_space(3))) v4i_gcc as3_v4i;

__device__ __forceinline__ void async_ld16(const v4f* g, const v4f* l) {
    // On amdgcn, a generic pointer to __shared__ carries the LDS byte offset
    // in its low 32 bits (flat->LDS truncation), and a generic global pointer
    // equals the AS1 address, so integer casts produce valid AS pointers.
    __builtin_amdgcn_global_load_async_to_lds_b128(
        (as1_v4i*)(uintptr_t)g,
        (as3_v4i*)(uint32_t)(uintptr_t)l,
        0, 0);
}

__device__ __forceinline__ void wait_async_le2() {
#if __has_builtin(__builtin_amdgcn_s_wait_asynccnt)
    __builtin_amdgcn_s_wait_asynccnt(2);
#else
    asm volatile("s_wait_asynccnt 0x2" ::: "memory");
#endif
    asm volatile("" ::: "memory"); // keep LDS reads below the wait
}

__device__ __forceinline__ void wait_async_le0() {
#if __has_builtin(__builtin_amdgcn_s_wait_asynccnt)
    __builtin_amdgcn_s_wait_asynccnt(0);
#else
    asm volatile("s_wait_asynccnt 0x0" ::: "memory");
#endif
    asm volatile("" ::: "memory");
}
#endif // USE_ASYNC

__global__ __launch_bounds__(256) void gru_dirscan_kernel(
    const v4f* __restrict__ z,
    const v4f* __restrict__ h,
    const v4f* __restrict__ h0,
    v4f* __restrict__ out)
{
    const unsigned t   = threadIdx.x;
    const unsigned col = blockIdx.x * blockDim.x + t;  // 0..KCOLS-1
    const unsigned c   = col >> 12;                    // col / KHW4
    const unsigned s   = col & (KHW4 - 1);
    const size_t base  = (size_t)c * (KD * KHW4) + s;

    v4f hp = h0[(size_t)c * KHW4 + s];
    size_t idx = base + (size_t)(KD - 1) * KHW4;       // d = D-1

#if USE_ASYNC
    // Double-buffered async staging: 2 stages x {z,h} x 256 lanes x 16B = 16 KB.
    __shared__ v4f zbuf[2][256];
    __shared__ v4f hbuf[2][256];

    // Prime stage d=63 -> buf0, d=62 -> buf1 (2 async ops per stage per wave).
    async_ld16(&z[idx],        &zbuf[0][t]);
    async_ld16(&h[idx],        &hbuf[0][t]);
    async_ld16(&z[idx - KHW4], &zbuf[1][t]);
    async_ld16(&h[idx - KHW4], &hbuf[1][t]);

    unsigned p = 0;
    for (int d = KD - 1; d >= 1; --d) {
        // Async loads complete in order: asynccnt<=2 => stage d is in LDS.
        wait_async_le2();
        v4f zt = zbuf[p][t];
        v4f ht = hbuf[p][t];

        v4f r;
        r.x = zt.x * ht.x + (1.0f - zt.x) * hp.x;
        r.y = zt.y * ht.y + (1.0f - zt.y) * hp.y;
        r.z = zt.z * ht.z + (1.0f - zt.z) * hp.z;
        r.w = zt.w * ht.w + (1.0f - zt.w) * hp.w;
        hp = r;

        if (d >= 2) {
            // WAR guard: pending ds_loads of buf[p] must retire before the
            // async engine may overwrite it.
            asm volatile("s_wait_dscnt 0x0" ::: "memory");
            const size_t i2 = idx - 2 * (size_t)KHW4;   // stage d-2
            async_ld16(&z[i2], &zbuf[p][t]);
            async_ld16(&h[i2], &hbuf[p][t]);
        }

        __builtin_nontemporal_store(hp, &out[idx]);
        idx -= KHW4;
        p ^= 1u;
    }

    // d == 0: everything issued; drain.
    wait_async_le0();
    {
        v4f zt = zbuf[p][t];
        v4f ht = hbuf[p][t];
        v4f r;
        r.x = zt.x * ht.x + (1.0f - zt.x) * hp.x;
        r.y = zt.y * ht.y + (1.0f - zt.y) * hp.y;
        r.z = zt.z * ht.z + (1.0f - zt.z) * hp.z;
        r.w = zt.w * ht.w + (1.0f - zt.w) * hp.w;
        __builtin_nontemporal_store(r, &out[idx]);
    }
#else
    // Fallback: register-pipelined NT streaming scan.
    v4f zt = __builtin_nontemporal_load(&z[idx]);
    v4f ht = __builtin_nontemporal_load(&h[idx]);

    #pragma unroll 2
    for (int d = KD - 1; d > 0; --d) {
        const size_t nidx = idx - KHW4;
        if (d > 1) {
            __builtin_prefetch(&z[nidx - KHW4], 0, 0);
            __builtin_prefetch(&h[nidx - KHW4], 0, 0);
        }
        v4f zn = __builtin_nontemporal_load(&z[nidx]);
        v4f hn = __builtin_nontemporal_load(&h[nidx]);

        v4f r;
        r.x = zt.x * ht.x + (1.0f - zt.x) * hp.x;
        r.y = zt.y * ht.y + (1.0f - zt.y) * hp.y;
        r.z = zt.z * ht.z + (1.0f - zt.z) * hp.z;
        r.w = zt.w * ht.w + (1.0f - zt.w) * hp.w;
        hp = r;

        __builtin_nontemporal_store(hp, &out[idx]);
        idx = nidx; zt = zn; ht = hn;
    }

    v4f r;
    r.x = zt.x * ht.x + (1.0f - zt.x) * hp.x;
    r.y = zt.y * ht.y + (1.0f - zt.y) * hp.y;
    r.z = zt.z * ht.z + (1.0f - zt.z) * hp.z;
    r.w = zt.w * ht.w + (1.0f - zt.w) * hp.w;
    __builtin_nontemporal_store(r, &out[idx]);
#endif
}

extern "C" void kernel_launch(void* const* d_in, const int* in_sizes, int n_in,
                              void* d_out, int out_size, void* d_ws, size_t ws_size,
                              hipStream_t stream) {
    (void)in_sizes; (void)n_in; (void)d_ws; (void)ws_size; (void)out_size;

    const v4f* z  = (const v4f*)d_in[0];   // (1,32,64,128,128) fp32
    const v4f* h  = (const v4f*)d_in[1];   // (1,32,64,128,128) fp32
    const v4f* h0 = (const v4f*)d_in[2];   // (1,32,1,128,128)  fp32
    v4f* out      = (v4f*)d_out;           // (1,32,64,128,128) fp32

    const int threads = 256;
    const int blocks  = KCOLS / threads;   // 512 blocks, 4096 wave32s

    gru_dirscan_kernel<<<blocks, threads, 0, stream>>>(z, h, h0, out);
}